// NSMLayer_77627238908016
// MI455X (gfx1250) — compile-verified
//
#include <hip/hip_runtime.h>

// ---------------- problem constants ----------------
#define HH 128
#define WW 128
#define KK 9
#define NANCH (HH * WW * KK)        // 147456
#define NSORT 262144                // 2^18 >= NANCH
#define PRE_NMS_K 6000
#define NPAD 6144                   // 24 * 256, >= PRE_NMS_K
#define WORDS 192                   // NPAD / 32
#define NW_ACT 188                  // ceil(PRE_NMS_K / 32)
#define NUM_ROIS 300
#define IOU_THR 0.7f
#define SCORE_THR 0.05f
#define IMGSZ 1024.0f
#define CHUNK 256                   // column boxes staged per LDS chunk

typedef __attribute__((ext_vector_type(2))) float v2f;
typedef __attribute__((ext_vector_type(8))) float v8f;
typedef int v4i_vs __attribute__((__vector_size__(16)));  // matches builtin proto
typedef __attribute__((address_space(1))) v4i_vs* g_v4i_p;
typedef __attribute__((address_space(3))) v4i_vs* l_v4i_p;

#if defined(__has_builtin)
#if __has_builtin(__builtin_amdgcn_wmma_f32_16x16x4_f32)
#define HAVE_WMMA4 1
#endif
#if __has_builtin(__builtin_amdgcn_global_load_async_to_lds_b128) && \
    __has_builtin(__builtin_amdgcn_s_wait_asynccnt)
#define HAVE_ASYNC_LDS 1
#endif
#endif

// ---------------- 1) decode + key pack ----------------
__global__ __launch_bounds__(256) void decode_kernel(
    const float* __restrict__ x, const float* __restrict__ anc,
    float4* __restrict__ boxes, unsigned long long* __restrict__ keys) {
  int idx = blockIdx.x * 256 + threadIdx.x;
  if (idx >= NSORT) return;
  if (idx < NANCH) {
    const float* e = x + (size_t)idx * 6;
    float e0 = e[0], e1 = e[1], dy = e[2], dx = e[3], dh = e[4], dw = e[5];
    float score = 1.0f / (1.0f + expf(e1 - e0));   // softmax(e0,e1)[0]
    const float* a = anc + (size_t)idx * 4;
    float ah = a[2] - a[0], aw = a[3] - a[1];
    float acy = a[0] + 0.5f * ah, acx = a[1] + 0.5f * aw;
    float cy = acy + dy * ah, cx = acx + dx * aw;
    float bh = ah * expf(dh), bw = aw * expf(dw);
    float y1 = fminf(fmaxf(cy - 0.5f * bh, 0.0f), IMGSZ);
    float x1 = fminf(fmaxf(cx - 0.5f * bw, 0.0f), IMGSZ);
    float y2 = fminf(fmaxf(cy + 0.5f * bh, 0.0f), IMGSZ);
    float x2 = fminf(fmaxf(cx + 0.5f * bw, 0.0f), IMGSZ);
    boxes[idx] = make_float4(y1, x1, y2, x2);
    unsigned sb = __float_as_uint(score);  // positive float: bits order == value order
    keys[idx] = ((unsigned long long)sb << 32) |
                (unsigned long long)(0xFFFFFFFFu - (unsigned)idx);
  } else {
    keys[idx] = (unsigned long long)(0xFFFFFFFFu - (unsigned)idx);  // pad: score 0
  }
}

// ---------------- 2) bitonic sort (descending by key) ----------------
__global__ __launch_bounds__(256) void bitonic_global(
    unsigned long long* __restrict__ keys, int j, int k) {
  int tid = blockIdx.x * 256 + threadIdx.x;            // NSORT/2 threads
  int i = ((tid & ~(j - 1)) << 1) | (tid & (j - 1));   // bit j of i is clear
  int ixj = i | j;
  unsigned long long a = keys[i], b = keys[ixj];
  bool up = (i & k) != 0;                // descending overall when (i&k)==0
  bool sw = up ? (a > b) : (a < b);
  if (sw) { keys[i] = b; keys[ixj] = a; }
}

__global__ __launch_bounds__(512) void bitonic_shared(
    unsigned long long* __restrict__ keys, int jstart, int k) {
  __shared__ unsigned long long s[1024];
  int base = blockIdx.x * 1024;
  s[threadIdx.x] = keys[base + threadIdx.x];
  s[threadIdx.x + 512] = keys[base + threadIdx.x + 512];
  __syncthreads();
  for (int j = jstart; j >= 1; j >>= 1) {
    int tid = threadIdx.x;
    int i = ((tid & ~(j - 1)) << 1) | (tid & (j - 1));
    int ixj = i | j;
    unsigned long long a = s[i], b = s[ixj];
    bool up = ((base + i) & k) != 0;
    bool sw = up ? (a > b) : (a < b);
    if (sw) { s[i] = b; s[ixj] = a; }
    __syncthreads();
  }
  keys[base + threadIdx.x] = s[threadIdx.x];
  keys[base + threadIdx.x + 512] = s[threadIdx.x + 512];
}

// ---------------- 3) gather top-K boxes, areas, score-threshold ----------------
__global__ __launch_bounds__(256) void gather_kernel(
    const unsigned long long* __restrict__ keys, const float4* __restrict__ boxes,
    float4* __restrict__ topb, float* __restrict__ tarea,
    unsigned* __restrict__ keep0) {
  int t = blockIdx.x * 256 + threadIdx.x;
  if (t >= NPAD) return;
  if (t < PRE_NMS_K) {
    unsigned long long kv = keys[t];
    unsigned idx = 0xFFFFFFFFu - (unsigned)(kv & 0xFFFFFFFFu);
    float score = __uint_as_float((unsigned)(kv >> 32));
    float4 b = boxes[idx];
    topb[t] = b;
    tarea[t] = (b.z - b.x) * (b.w - b.y);
    keep0[t] = (score > SCORE_THR) ? 1u : 0u;
  } else {
    topb[t] = make_float4(0.f, 0.f, 0.f, 0.f);  // sentinel: IoU with anything = 0
    tarea[t] = 0.f;
    keep0[t] = 0u;
  }
}

// ---------------- 4) pairwise IoU -> suppression bitmask ----------------
// Block = 128 threads (4 waves). Wave w owns 16 rows; block owns 64 rows.
// area_i + area_j for each 16x16 tile via V_WMMA_F32_16X16X4_F32:
//   A (16x4): col0 = area_row, col1 = 1        -> lane<16: {area,1}, lane>=16: {0,0}
//   B (4x16): row0 = 1,        row1 = area_col -> lane<16: {1,area}, lane>=16: {0,0}
// D[m][n] = area_row[m] + area_col[n], in the standard 16x16 f32 C/D layout.
__global__ __launch_bounds__(128) void iou_mask_kernel(
    const float4* __restrict__ topb, const float* __restrict__ tarea,
    unsigned* __restrict__ supp) {
  __shared__ float4 s_cbox[CHUNK];
  __shared__ float s_carea[CHUNK];
  __shared__ float4 s_rbox[64];
  __shared__ float s_rarea[64];
  int t = threadIdx.x;
  int lane = t & 31;
  int wave = t >> 5;
  int row0 = blockIdx.x * 64;
  if (t < 64) {
    s_rbox[t] = topb[row0 + t];
    s_rarea[t] = tarea[row0 + t];
  }
  __syncthreads();

  int i0 = row0 + wave * 16;
  float ra = s_rarea[wave * 16 + (lane & 15)];
  v2f afrag;
  afrag.x = (lane < 16) ? ra : 0.0f;
  afrag.y = (lane < 16) ? 1.0f : 0.0f;

  for (int cc = 0; cc < NPAD / CHUNK; ++cc) {
    __syncthreads();  // previous chunk fully consumed
    int j0 = cc * CHUNK;
    for (int q = t; q < CHUNK; q += 128) {
#if HAVE_ASYNC_LDS
      __builtin_amdgcn_global_load_async_to_lds_b128(
          (g_v4i_p)(topb + j0 + q), (l_v4i_p)(&s_cbox[q]), 0, 0);
#else
      s_cbox[q] = topb[j0 + q];
#endif
      s_carea[q] = tarea[j0 + q];
    }
#if HAVE_ASYNC_LDS
    __builtin_amdgcn_s_wait_asynccnt(0);
#endif
    __syncthreads();

    for (int g = 0; g < 8; ++g) {          // 8 groups of 32 columns per chunk
      int cbase = g * 32;
      float caA = s_carea[cbase + (lane & 15)];
      float caB = s_carea[cbase + 16 + (lane & 15)];
      v2f bfA, bfB;
      bfA.x = (lane < 16) ? 1.0f : 0.0f;
      bfA.y = (lane < 16) ? caA : 0.0f;
      bfB.x = (lane < 16) ? 1.0f : 0.0f;
      bfB.y = (lane < 16) ? caB : 0.0f;
#if HAVE_WMMA4
      v8f czero = {};
      v8f sA = __builtin_amdgcn_wmma_f32_16x16x4_f32(
          false, afrag, false, bfA, (short)0, czero, false, false);
      v8f sB = __builtin_amdgcn_wmma_f32_16x16x4_f32(
          false, afrag, false, bfB, (short)0, czero, false, false);
#endif
      float4 cbA = s_cbox[cbase + (lane & 15)];
      float4 cbB = s_cbox[cbase + 16 + (lane & 15)];
      unsigned myword = 0u;
#pragma unroll
      for (int r = 0; r < 8; ++r) {
        int rrow = wave * 16 + ((lane < 16) ? r : (r + 8));
        float4 rb = s_rbox[rrow];
        // tile A (cols cbase .. cbase+15)
        float iy1 = fmaxf(rb.x, cbA.x), ix1 = fmaxf(rb.y, cbA.y);
        float iy2 = fminf(rb.z, cbA.z), ix2 = fminf(rb.w, cbA.w);
        float interA = fmaxf(iy2 - iy1, 0.f) * fmaxf(ix2 - ix1, 0.f);
#if HAVE_WMMA4
        float areasumA = sA[r];
#else
        float areasumA = s_rarea[rrow] + caA;
#endif
        float unionA = areasumA - interA + 1e-9f;
        bool pA = interA > IOU_THR * unionA;   // iou > thr, union > 0
        // tile B (cols cbase+16 .. cbase+31)
        iy1 = fmaxf(rb.x, cbB.x); ix1 = fmaxf(rb.y, cbB.y);
        iy2 = fminf(rb.z, cbB.z); ix2 = fminf(rb.w, cbB.w);
        float interB = fmaxf(iy2 - iy1, 0.f) * fmaxf(ix2 - ix1, 0.f);
#if HAVE_WMMA4
        float areasumB = sB[r];
#else
        float areasumB = s_rarea[rrow] + caB;
#endif
        float unionB = areasumB - interB + 1e-9f;
        bool pB = interB > IOU_THR * unionB;

        unsigned balA = __builtin_amdgcn_ballot_w32(pA);
        unsigned balB = __builtin_amdgcn_ballot_w32(pB);
        // balA low16 = row r cols [0..15]; high16 = row r+8 cols [0..15]
        unsigned wlo = (balA & 0xFFFFu) | (balB << 16);          // row r, 32 cols
        unsigned whi = (balA >> 16) | (balB & 0xFFFF0000u);      // row r+8
        if (lane == r) myword = wlo;
        if (lane == r + 8) myword = whi;
      }
      if (lane < 16) {
        int row = i0 + lane;
        supp[(size_t)row * WORDS + (unsigned)((j0 + cbase) >> 5)] = myword;
      }
    }
  }
}

// ---------------- 5) greedy NMS reduction (one workgroup) ----------------
__global__ __launch_bounds__(WORDS) void nms_reduce_kernel(
    const unsigned* __restrict__ supp, const unsigned* __restrict__ keep0,
    unsigned* __restrict__ keepw_out) {
  __shared__ unsigned keepw[WORDS];
  __shared__ unsigned diag[32];
  __shared__ unsigned kcur;
  int t = threadIdx.x;  // 0..191
  unsigned w = 0;
  for (int b = 0; b < 32; ++b) w |= (keep0[t * 32 + b] ? 1u : 0u) << b;
  keepw[t] = w;
  __syncthreads();
  for (int wi = 0; wi < NW_ACT; ++wi) {
    if (t < 32) diag[t] = supp[(size_t)(wi * 32 + t) * WORDS + wi];
    __syncthreads();
    if (t == 0) {
      unsigned kw = keepw[wi];
      for (int b = 0; b < 32; ++b) {
        if ((kw >> b) & 1u) {
          unsigned m = (b == 31) ? 0u : (0xFFFFFFFFu << (b + 1));  // bits > b
          kw &= ~(diag[b] & m);
        }
      }
      keepw[wi] = kw;
      kcur = kw;
    }
    __syncthreads();
    unsigned kw = kcur;
    if (t > wi) {
      unsigned mine = keepw[t];
      unsigned kk = kw;
      while (kk) {
        int b = __builtin_ctz(kk);
        kk &= kk - 1;
        mine &= ~supp[(size_t)(wi * 32 + b) * WORDS + t];
      }
      keepw[t] = mine;
    }
    __syncthreads();
  }
  keepw_out[t] = keepw[t];
}

// ---------------- 6) compact kept boxes -> 300 ROIs (zero padded) ----------------
__global__ __launch_bounds__(WORDS) void emit_kernel(
    const unsigned* __restrict__ keepw, const float4* __restrict__ topb,
    float* __restrict__ out) {
  __shared__ unsigned base[WORDS];
  __shared__ unsigned pc[WORDS];
  int t = threadIdx.x;
  for (int q = t; q < NUM_ROIS * 4; q += WORDS) out[q] = 0.0f;
  unsigned kw = keepw[t];
  pc[t] = (unsigned)__builtin_popcount(kw);
  __syncthreads();
  if (t == 0) {
    unsigned s = 0;
    for (int i = 0; i < WORDS; ++i) { base[i] = s; s += pc[i]; }
  }
  __syncthreads();
  unsigned pos = base[t];
  while (kw) {
    int b = __builtin_ctz(kw);
    kw &= kw - 1;
    if (pos < NUM_ROIS) {
      float4 bx = topb[t * 32 + b];
      out[pos * 4 + 0] = bx.x;
      out[pos * 4 + 1] = bx.y;
      out[pos * 4 + 2] = bx.z;
      out[pos * 4 + 3] = bx.w;
    }
    ++pos;
  }
}

// ---------------- launch ----------------
extern "C" void kernel_launch(void* const* d_in, const int* in_sizes, int n_in,
                              void* d_out, int out_size, void* d_ws, size_t ws_size,
                              hipStream_t stream) {
  (void)in_sizes; (void)n_in; (void)out_size; (void)ws_size;
  const float* x = (const float*)d_in[0];
  const float* anc = (const float*)d_in[1];
  float* out = (float*)d_out;

  char* ws = (char*)d_ws;
  size_t off = 0;
  auto take = [&](size_t bytes) -> void* {
    void* p = ws + off;
    off = (off + bytes + 255) & ~(size_t)255;
    return p;
  };
  unsigned long long* keys = (unsigned long long*)take((size_t)NSORT * 8);
  float4* boxes = (float4*)take((size_t)NANCH * 16);
  float4* topb = (float4*)take((size_t)NPAD * 16);
  float* tarea = (float*)take((size_t)NPAD * 4);
  unsigned* keep0 = (unsigned*)take((size_t)NPAD * 4);
  unsigned* keepw = (unsigned*)take((size_t)WORDS * 4);
  unsigned* supp = (unsigned*)take((size_t)NPAD * WORDS * 4);

  decode_kernel<<<NSORT / 256, 256, 0, stream>>>(x, anc, boxes, keys);

  for (int k = 2; k <= NSORT; k <<= 1) {
    for (int j = k >> 1; j >= 1024; j >>= 1)
      bitonic_global<<<NSORT / 2 / 256, 256, 0, stream>>>(keys, j, k);
    int jstart = ((k >> 1) < 512) ? (k >> 1) : 512;
    bitonic_shared<<<NSORT / 1024, 512, 0, stream>>>(keys, jstart, k);
  }

  gather_kernel<<<NPAD / 256, 256, 0, stream>>>(keys, boxes, topb, tarea, keep0);
  iou_mask_kernel<<<NPAD / 64, 128, 0, stream>>>(topb, tarea, supp);
  nms_reduce_kernel<<<1, WORDS, 0, stream>>>(supp, keep0, keepw);
  emit_kernel<<<1, WORDS, 0, stream>>>(keepw, topb, out);
}